// ParametricLinearSystem_2671469658335
// MI455X (gfx1250) — compile-verified
//
#include <hip/hip_runtime.h>

// ---------------------------------------------------------------------------
// Linear SSM via chunked parallel scan, all matmuls on v_wmma_f32_16x16x4_f32.
//   x(t+1) = A x(t) + B u(t);  y[t] = C x(t+1) + D u(t) = (CA) x(t) + (CB+D) u(t)
// Phase 0: P=A^64, CA=C*A, E=C*B+D          (1 WG, LDS ping-pong squaring)
// Phase 1: per chunk c: S_c = local terminal state, zero init (parallel)
// Phase 2: Z_{c+1} = Z_c P^T + S_c          (serial over 64 chunks, tiny)
// Phase 3: per chunk: exact recurrence from Z_c, emit y      (parallel)
// u tiles for the next timestep are staged into LDS with
// GLOBAL_LOAD_ASYNC_TO_LDS_B64 (ASYNCcnt), double-buffered against compute.
// ---------------------------------------------------------------------------

typedef float v2f __attribute__((ext_vector_type(2)));
typedef float v8f __attribute__((ext_vector_type(8)));

#define BSZ    32
#define TLEN   4096
#define DIM    128
#define CHUNK  64
#define NCHUNK (TLEN / CHUNK)

// workspace layout in floats
#define WS_P   0
#define WS_CA  16384
#define WS_E   32768
#define WS_S   49152                         // [NCHUNK][32][128]
#define WS_Z   (49152 + NCHUNK * BSZ * DIM)  // [NCHUNK][32][128]

static __device__ __forceinline__ v8f wmma4(v2f a, v2f b, v8f c) {
  // D = A(16x4 f32) * B(4x16 f32) + C(16x16 f32)
  return __builtin_amdgcn_wmma_f32_16x16x4_f32(false, a, false, b,
                                               (short)0, c, false, false);
}

// async copy 8 bytes global -> LDS (tracked by ASYNCcnt)
static __device__ __forceinline__ void async_ld_b64(unsigned lds_off,
                                                    const void* gptr) {
  asm volatile("global_load_async_to_lds_b64 %0, %1, off"
               :: "v"(lds_off), "v"(gptr)
               : "memory");
}
static __device__ __forceinline__ void wait_async0() {
  asm volatile("s_wait_asynccnt 0x0" ::: "memory");
}
// flat pointer -> LDS byte offset (low 32 bits of the flat address)
static __device__ __forceinline__ unsigned lds_off32(const void* p) {
  return (unsigned)(size_t)p;
}

// ---------------------------------------------------------------------------
// K0: CA = C*A, E = C*B + D, P = A^64 (6 squarings in LDS ping-pong)
// ---------------------------------------------------------------------------
__global__ __launch_bounds__(256) void k0_precompute(
    const float* __restrict__ A, const float* __restrict__ Bm,
    const float* __restrict__ Cm, const float* __restrict__ Dm,
    float* __restrict__ ws) {
  __shared__ float sm0[DIM * DIM];
  __shared__ float sm1[DIM * DIM];
  const int tid = threadIdx.x;
  const int lane = tid & 31, wave = tid >> 5;
  const int l16 = lane & 15, h = lane >> 4;
  float* P  = ws + WS_P;
  float* CA = ws + WS_CA;
  float* E  = ws + WS_E;

  // ---- CA = C*A and E = C*B + D (each wave owns one n-tile, loops m-tiles)
  for (int mt = 0; mt < 8; ++mt) {
    const int m0 = mt * 16, n0 = wave * 16;
    v8f acc0 = {};
    v8f acc1;
    #pragma unroll
    for (int r = 0; r < 8; ++r)
      acc1[r] = Dm[(m0 + r + 8 * h) * DIM + n0 + l16];
    #pragma unroll
    for (int kt = 0; kt < 32; ++kt) {
      const int k0 = kt * 4 + 2 * h;
      v2f cf = *(const v2f*)&Cm[(m0 + l16) * DIM + k0];
      v2f af; af.x = A [k0 * DIM + n0 + l16]; af.y = A [(k0 + 1) * DIM + n0 + l16];
      v2f bf; bf.x = Bm[k0 * DIM + n0 + l16]; bf.y = Bm[(k0 + 1) * DIM + n0 + l16];
      acc0 = wmma4(cf, af, acc0);
      acc1 = wmma4(cf, bf, acc1);
    }
    #pragma unroll
    for (int r = 0; r < 8; ++r) {
      CA[(m0 + r + 8 * h) * DIM + n0 + l16] = acc0[r];
      E [(m0 + r + 8 * h) * DIM + n0 + l16] = acc1[r];
    }
  }

  // ---- P = A^64 via 6 squarings
  for (int i = tid; i < DIM * DIM; i += 256) sm0[i] = A[i];
  __syncthreads();
  for (int s = 0; s < 6; ++s) {
    const float* src = (s & 1) ? sm1 : sm0;
    float*       dst = (s & 1) ? sm0 : sm1;
    for (int mt = 0; mt < 8; ++mt) {
      const int m0 = mt * 16, n0 = wave * 16;
      v8f acc = {};
      #pragma unroll
      for (int kt = 0; kt < 32; ++kt) {
        const int k0 = kt * 4 + 2 * h;
        v2f a = *(const v2f*)&src[(m0 + l16) * DIM + k0];
        v2f b; b.x = src[k0 * DIM + n0 + l16]; b.y = src[(k0 + 1) * DIM + n0 + l16];
        acc = wmma4(a, b, acc);
      }
      #pragma unroll
      for (int r = 0; r < 8; ++r)
        dst[(m0 + r + 8 * h) * DIM + n0 + l16] = acc[r];
    }
    __syncthreads();
  }
  // after 6 squarings result sits in sm0
  for (int i = tid; i < DIM * DIM; i += 256) P[i] = sm0[i];
}

// ---------------------------------------------------------------------------
// K1: per (chunk, batch-tile): S_c = terminal local state with zero init
//     256 threads / 8 waves; u staged via async-to-LDS double buffer
// ---------------------------------------------------------------------------
__global__ __launch_bounds__(256) void k1_localstate(
    const float* __restrict__ U, const float* __restrict__ A,
    const float* __restrict__ Bm, float* __restrict__ ws) {
  __shared__ float xbuf[2][16 * DIM];
  __shared__ float ubuf[2][16 * DIM];
  const int c = blockIdx.x >> 1, bt = blockIdx.x & 1, b0 = bt * 16;
  const int tid = threadIdx.x, lane = tid & 31, wave = tid >> 5;
  const int l16 = lane & 15, h = lane >> 4;
  const int n0 = wave * 16;

  v2f Af[32], Bf[32];
  #pragma unroll
  for (int kt = 0; kt < 32; ++kt) {
    const int col = kt * 4 + 2 * h;
    Af[kt] = *(const v2f*)&A [(n0 + l16) * DIM + col];
    Bf[kt] = *(const v2f*)&Bm[(n0 + l16) * DIM + col];
  }
  for (int i = tid; i < 16 * DIM; i += 256) xbuf[0][i] = 0.0f;

  // stage u(t = c*CHUNK) into ubuf[0]: 2048 floats, 2 per thread per issue
  {
    const int t0 = c * CHUNK;
    #pragma unroll
    for (int p = 0; p < 4; ++p) {
      const int i = p * 512 + tid * 2;
      const int row = i >> 7, col = i & 127;
      async_ld_b64(lds_off32(&ubuf[0][i]),
                   U + ((size_t)(b0 + row) * TLEN + t0) * DIM + col);
    }
  }
  wait_async0();
  __syncthreads();

  for (int s = 0; s < CHUNK; ++s) {
    const int t = c * CHUNK + s;
    const float* xin = xbuf[s & 1];
    const float* uin = ubuf[s & 1];
    float* xout = xbuf[(s + 1) & 1];
    if (s + 1 < CHUNK) {  // stage next u tile while computing
      #pragma unroll
      for (int p = 0; p < 4; ++p) {
        const int i = p * 512 + tid * 2;
        const int row = i >> 7, col = i & 127;
        async_ld_b64(lds_off32(&ubuf[(s + 1) & 1][i]),
                     U + ((size_t)(b0 + row) * TLEN + t + 1) * DIM + col);
      }
    }
    v8f accA = {}, accB = {};
    #pragma unroll
    for (int kt = 0; kt < 32; ++kt) {
      const int o = kt * 4 + 2 * h;
      v2f xf = *(const v2f*)&xin[l16 * DIM + o];
      v2f uf = *(const v2f*)&uin[l16 * DIM + o];
      accA = wmma4(xf, Af[kt], accA);
      accB = wmma4(uf, Bf[kt], accB);
    }
    #pragma unroll
    for (int r = 0; r < 8; ++r)
      xout[(r + 8 * h) * DIM + n0 + l16] = accA[r] + accB[r];
    wait_async0();
    __syncthreads();
  }
  // CHUNK is even -> final state in xbuf[0]
  float* S = ws + WS_S + c * (BSZ * DIM) + b0 * DIM;
  for (int i = tid; i < 16 * DIM; i += 256) S[i] = xbuf[0][i];
}

// ---------------------------------------------------------------------------
// K2: serial chunk scan  Z_{c+1} = Z_c * P^T + S_c   (1 WG, 16 waves)
// ---------------------------------------------------------------------------
__global__ __launch_bounds__(512) void k2_scan(float* __restrict__ ws) {
  __shared__ float zbuf[2][BSZ * DIM];
  const int tid = threadIdx.x, lane = tid & 31, wave = tid >> 5;
  const int l16 = lane & 15, h = lane >> 4;
  const int mt = wave >> 3, nt = wave & 7;
  const int m0 = mt * 16, n0 = nt * 16;
  const float* P = ws + WS_P;
  const float* S = ws + WS_S;
  float* Z = ws + WS_Z;

  v2f Pf[32];
  #pragma unroll
  for (int kt = 0; kt < 32; ++kt)
    Pf[kt] = *(const v2f*)&P[(n0 + l16) * DIM + kt * 4 + 2 * h];

  for (int i = tid; i < BSZ * DIM; i += 512) { zbuf[0][i] = 0.0f; Z[i] = 0.0f; }
  __syncthreads();

  for (int ci = 0; ci < NCHUNK - 1; ++ci) {
    const float* zin = zbuf[ci & 1];
    float* zout = zbuf[(ci + 1) & 1];
    v8f acc;
    #pragma unroll
    for (int r = 0; r < 8; ++r)
      acc[r] = S[ci * BSZ * DIM + (m0 + r + 8 * h) * DIM + n0 + l16];
    #pragma unroll
    for (int kt = 0; kt < 32; ++kt) {
      v2f zf = *(const v2f*)&zin[(m0 + l16) * DIM + kt * 4 + 2 * h];
      acc = wmma4(zf, Pf[kt], acc);
    }
    #pragma unroll
    for (int r = 0; r < 8; ++r) {
      const float v = acc[r];
      zout[(m0 + r + 8 * h) * DIM + n0 + l16] = v;
      Z[(ci + 1) * BSZ * DIM + (m0 + r + 8 * h) * DIM + n0 + l16] = v;
    }
    __syncthreads();
  }
}

// ---------------------------------------------------------------------------
// K3: per (chunk, batch-tile): exact recurrence from Z_c, emit y
//     512 threads / 16 waves: waves 0-7 hold A,B frags -> next state
//                             waves 8-15 hold CA,E frags -> y output
//     y[t] = CA*x(t) + E*u(t);  x(t+1) = A*x(t) + B*u(t)
// ---------------------------------------------------------------------------
__global__ __launch_bounds__(512) void k3_output(
    const float* __restrict__ U, const float* __restrict__ A,
    const float* __restrict__ Bm, const float* __restrict__ ws,
    float* __restrict__ out) {
  __shared__ float xbuf[2][16 * DIM];
  __shared__ float ubuf[2][16 * DIM];
  const int c = blockIdx.x >> 1, bt = blockIdx.x & 1, b0 = bt * 16;
  const int tid = threadIdx.x, lane = tid & 31, wave = tid >> 5;
  const int l16 = lane & 15, h = lane >> 4;
  const bool isY = wave >= 8;         // wave-uniform split
  const int n0 = (wave & 7) * 16;
  const float* Z = ws + WS_Z;

  // resident weight fragments: state waves {A, B}; y waves {CA, E}
  const float* W0 = isY ? (ws + WS_CA) : A;
  const float* W1 = isY ? (ws + WS_E)  : Bm;
  v2f F0[32], F1[32];
  #pragma unroll
  for (int kt = 0; kt < 32; ++kt) {
    const int col = kt * 4 + 2 * h;
    F0[kt] = *(const v2f*)&W0[(n0 + l16) * DIM + col];
    F1[kt] = *(const v2f*)&W1[(n0 + l16) * DIM + col];
  }
  // chunk entry state x(c*CHUNK) = Z_c (16 contiguous rows for this tile)
  for (int i = tid; i < 16 * DIM; i += 512)
    xbuf[0][i] = Z[c * BSZ * DIM + b0 * DIM + i];

  // stage u(t = c*CHUNK): 2048 floats, 2 per thread per issue, 512 threads
  {
    const int t0 = c * CHUNK;
    #pragma unroll
    for (int p = 0; p < 2; ++p) {
      const int i = p * 1024 + tid * 2;
      const int row = i >> 7, col = i & 127;
      async_ld_b64(lds_off32(&ubuf[0][i]),
                   U + ((size_t)(b0 + row) * TLEN + t0) * DIM + col);
    }
  }
  wait_async0();
  __syncthreads();

  for (int s = 0; s < CHUNK; ++s) {
    const int t = c * CHUNK + s;
    const float* xin = xbuf[s & 1];
    const float* uin = ubuf[s & 1];
    float* xout = xbuf[(s + 1) & 1];
    if (s + 1 < CHUNK) {  // stage next u tile while computing
      #pragma unroll
      for (int p = 0; p < 2; ++p) {
        const int i = p * 1024 + tid * 2;
        const int row = i >> 7, col = i & 127;
        async_ld_b64(lds_off32(&ubuf[(s + 1) & 1][i]),
                     U + ((size_t)(b0 + row) * TLEN + t + 1) * DIM + col);
      }
    }
    v8f acc0 = {}, acc1 = {};
    #pragma unroll
    for (int kt = 0; kt < 32; ++kt) {
      const int o = kt * 4 + 2 * h;
      v2f xf = *(const v2f*)&xin[l16 * DIM + o];
      v2f uf = *(const v2f*)&uin[l16 * DIM + o];
      acc0 = wmma4(xf, F0[kt], acc0);   // x*A^T      | x*(CA)^T
      acc1 = wmma4(uf, F1[kt], acc1);   // u*B^T      | u*E^T
    }
    if (isY) {
      #pragma unroll
      for (int r = 0; r < 8; ++r)
        out[((size_t)(b0 + r + 8 * h) * TLEN + t) * DIM + n0 + l16] =
            acc0[r] + acc1[r];
    } else {
      #pragma unroll
      for (int r = 0; r < 8; ++r)
        xout[(r + 8 * h) * DIM + n0 + l16] = acc0[r] + acc1[r];
    }
    wait_async0();
    __syncthreads();
  }
}

// ---------------------------------------------------------------------------
extern "C" void kernel_launch(void* const* d_in, const int* in_sizes, int n_in,
                              void* d_out, int out_size, void* d_ws,
                              size_t ws_size, hipStream_t stream) {
  (void)in_sizes; (void)n_in; (void)out_size; (void)ws_size;
  const float* U  = (const float*)d_in[0];
  const float* A  = (const float*)d_in[1];
  const float* Bm = (const float*)d_in[2];
  const float* Cm = (const float*)d_in[3];
  const float* Dm = (const float*)d_in[4];
  float* ws  = (float*)d_ws;
  float* out = (float*)d_out;

  k0_precompute<<<1, 256, 0, stream>>>(A, Bm, Cm, Dm, ws);
  k1_localstate<<<NCHUNK * 2, 256, 0, stream>>>(U, A, Bm, ws);
  k2_scan<<<1, 512, 0, stream>>>(ws);
  k3_output<<<NCHUNK * 2, 512, 0, stream>>>(U, A, Bm, ws, out);
}